// HistogramLayer_13048110645959
// MI455X (gfx1250) — compile-verified
//
#include <hip/hip_runtime.h>
#include <stdint.h>

// HistogramLayer inference bias:
//   bin(b,d)  = clip(searchsorted_right(edges[:,d], x[b,d]) - 1, 0, 7)
//             = sum_{j=1..7} [ x[b,d] >= edges[j,d] ]          (7 compares)
//   bias[b]   = prod_d freq[bin,d] / sum_b freq[b,d]
//             = (prod_d freq[bin(b,d), d]) * C,   C = 1 / prod_d sum_b freq[b,d]
//
// Memory-bound: 272 MB of traffic -> ~11.7 us floor @ 23.3 TB/s. CDNA5 path:
// async global->LDS DMA (ASYNCcnt) with double-buffered 16 KB tiles.

#define D_FEAT    16
#define NBINS     8
#define TPB       256     // 8 wave32 per block
#define ROWS_TILE 256     // one row (64 B) per thread per tile
#define TILE_CHUNKS (ROWS_TILE * D_FEAT * 4 / 16)   // 1024 x 16B chunks
#define CHUNKS_PER_THREAD (TILE_CHUNKS / TPB)       // 4

// ---- CDNA5 async global->LDS copy (inline asm: portable across toolchains) ----
__device__ __forceinline__ void async_copy_b128(uint32_t lds_byte_addr, const void* gptr) {
    asm volatile("global_load_async_to_lds_b128 %0, %1, off"
                 :
                 : "v"(lds_byte_addr), "v"((uint64_t)(uintptr_t)gptr)
                 : "memory");
}
__device__ __forceinline__ void wait_async_le4() {
    asm volatile("s_wait_asynccnt 0x4" ::: "memory");
}
__device__ __forceinline__ void wait_async_0() {
    asm volatile("s_wait_asynccnt 0x0" ::: "memory");
}

// generic (flat) pointer to a __shared__ object -> wave-relative LDS byte offset
__device__ __forceinline__ uint32_t lds_offset_of(const void* p) {
    return (uint32_t)(uintptr_t)p;   // LDS aperture: addr[31:0] is the LDS offset
}

__device__ __forceinline__ void stage_tile(uint32_t lds_base, const float* gbase, int t) {
#pragma unroll
    for (int k = 0; k < CHUNKS_PER_THREAD; ++k) {
        const int chunk = t + k * TPB;          // contiguous 16B chunks across the block
        async_copy_b128(lds_base + (uint32_t)chunk * 16u,
                        (const char*)gbase + (size_t)chunk * 16u);
    }
}

// compare/select ladder against uniform (SGPR-resident) edge & freq tables
__device__ __forceinline__ float row_bias(const float* __restrict__ freq,
                                          const float* __restrict__ edges,
                                          const float xv[D_FEAT], float C) {
    float prod = C;
#pragma unroll
    for (int d = 0; d < D_FEAT; ++d) {
        const float x = xv[d];
        float p = freq[0 * D_FEAT + d];
#pragma unroll
        for (int j = 1; j < NBINS; ++j)
            p = (x >= edges[j * D_FEAT + d]) ? freq[j * D_FEAT + d] : p;
        prod *= p;
    }
    return prod;
}

__global__ void __launch_bounds__(TPB)
hist_bias_kernel(const float* __restrict__ inp,
                 const float* __restrict__ freq,
                 const float* __restrict__ edges,
                 float* __restrict__ out,
                 int nrows) {
    __shared__ __align__(16) float tile0[ROWS_TILE * D_FEAT];   // 16 KB
    __shared__ __align__(16) float tile1[ROWS_TILE * D_FEAT];   // 16 KB

    const int t      = threadIdx.x;
    const int tiles  = nrows / ROWS_TILE;
    const int stride = gridDim.x;

    // Row-independent normalization constant (uniform math, freq in [1,2)).
    float csum = 1.0f;
#pragma unroll
    for (int d = 0; d < D_FEAT; ++d) {
        float s = 0.0f;
#pragma unroll
        for (int b = 0; b < NBINS; ++b) s += freq[b * D_FEAT + d];
        csum *= s;
    }
    const float C = 1.0f / csum;

    // NOTE: no aggregate initializers holding shared-memory pointers
    // (ld.lld cannot relocate addrspacecast in static initializers);
    // select the double-buffer via ternaries on an index instead.
    const uint32_t lds0 = lds_offset_of(tile0);
    const uint32_t lds1 = lds_offset_of(tile1);

    int cur  = 0;
    int tile = blockIdx.x;
    if (tile < tiles)
        stage_tile(lds0, inp + (size_t)tile * ROWS_TILE * D_FEAT, t);

    for (; tile < tiles; tile += stride) {
        const int nt = tile + stride;
        if (nt < tiles) {
            stage_tile((cur == 0) ? lds1 : lds0,
                       inp + (size_t)nt * ROWS_TILE * D_FEAT, t);
            wait_async_le4();    // async loads retire in order: current tile is done
        } else {
            wait_async_0();
        }
        __syncthreads();         // all waves' staging of current tile visible

        const float* buf = (cur == 0) ? tile0 : tile1;
        const float4* row4 = (const float4*)(buf + t * D_FEAT);
        const float4 v0 = row4[0], v1 = row4[1], v2 = row4[2], v3 = row4[3];
        const float xv[D_FEAT] = { v0.x, v0.y, v0.z, v0.w,
                                   v1.x, v1.y, v1.z, v1.w,
                                   v2.x, v2.y, v2.z, v2.w,
                                   v3.x, v3.y, v3.z, v3.w };
        out[(size_t)tile * ROWS_TILE + t] = row_bias(freq, edges, xv, C);

        __syncthreads();         // done reading before this buffer is re-staged
        cur ^= 1;
    }

    // Tail rows (nrows not a multiple of ROWS_TILE): direct global path.
    for (int r = tiles * ROWS_TILE + blockIdx.x * TPB + t; r < nrows;
         r += stride * TPB) {
        const float* g = inp + (size_t)r * D_FEAT;
        float xv[D_FEAT];
#pragma unroll
        for (int d = 0; d < D_FEAT; ++d) xv[d] = g[d];
        out[r] = row_bias(freq, edges, xv, C);
    }
}

extern "C" void kernel_launch(void* const* d_in, const int* in_sizes, int n_in,
                              void* d_out, int out_size, void* d_ws, size_t ws_size,
                              hipStream_t stream) {
    const float* inp   = (const float*)d_in[0];   // [B, 16] fp32
    const float* freq  = (const float*)d_in[1];   // [8, 16] fp32
    const float* edges = (const float*)d_in[2];   // [9, 16] fp32
    float*       out   = (float*)d_out;           // [B] fp32

    const int nrows = in_sizes[0] / D_FEAT;
    int tiles  = nrows / ROWS_TILE;
    int blocks = tiles;
    if (blocks > 4096) blocks = 4096;   // persistent blocks: double-buffer pipeline
    if (blocks < 1)    blocks = 1;

    hist_bias_kernel<<<dim3(blocks), dim3(TPB), 0, stream>>>(inp, freq, edges, out, nrows);
}